// gca_uot_19902878449981
// MI455X (gfx1250) — compile-verified
//
#include <hip/hip_runtime.h>
#include <math.h>

#define NN 4096
#define DD 256
#define INV_EPS 20.0f
#define EPSC 0.05f
#define F1C (1e-4f/(1e-4f+0.05f))
#define EPSREL (0.05f+1e-4f)
#define AW (1.0f/4096.0f)
#define TAUC 1e5f
#define QC 0.6f
#define LAMC 0.01f
#define R1C 1.0f
#define R2C 1.0f

typedef float v2f __attribute__((ext_vector_type(2)));
typedef float v8f __attribute__((ext_vector_type(8)));

// scalar flag slots in workspace
#define S_CMAX 0
#define S_MAXU 1
#define S_MAXV 2
#define S_NONF 3
#define S_BAD  4
#define S_DONE 5

// ---------------- init ----------------
__global__ __launch_bounds__(256) void k_init(float* u, float* v, float* f, float* g,
                                              unsigned* S) {
  int j = blockIdx.x * 256 + threadIdx.x;
  u[j] = AW; v[j] = AW; f[j] = 0.0f; g[j] = 0.0f;
  if (blockIdx.x == 0 && threadIdx.x < 8) S[threadIdx.x] = 0u;
}

// ---------------- row normalize ----------------
__global__ __launch_bounds__(256) void k_rownorm(const float* __restrict__ z,
                                                 float* __restrict__ zn) {
  int i = blockIdx.x, t = threadIdx.x;
  float x = z[(size_t)i * DD + t];
  __shared__ float sm[256];
  sm[t] = x * x; __syncthreads();
  for (int s = 128; s > 0; s >>= 1) { if (t < s) sm[t] += sm[t + s]; __syncthreads(); }
  float nrm = sqrtf(sm[0]);
  zn[(size_t)i * DD + t] = x / nrm;
}

// ---------------- cost GEMM via f32 WMMA (exact) ----------------
// block = 8 waves; wave computes a 32x32 tile; block tile = 64 rows x 128 cols.
__global__ __launch_bounds__(256) void k_gemm_cost(const float* __restrict__ Z,
                                                   float* __restrict__ C,
                                                   unsigned* S) {
  const int lane = threadIdx.x & 31;
  const int wid  = threadIdx.x >> 5;
  const int m  = lane & 15;
  const int hi = lane >> 4;
  const int rb = blockIdx.y * 64  + (wid & 1) * 32;
  const int cb = blockIdx.x * 128 + (wid >> 1) * 32;
  const float* pa0 = Z + (size_t)(rb + m) * DD;
  const float* pa1 = Z + (size_t)(rb + 16 + m) * DD;
  const float* pb0 = Z + (size_t)(cb + m) * DD;
  const float* pb1 = Z + (size_t)(cb + 16 + m) * DD;
  v8f c00 = {}, c01 = {}, c10 = {}, c11 = {};
#pragma unroll 4
  for (int kc = 0; kc < DD; kc += 4) {
    int ko = kc + hi * 2;  // A 16x4 f32 layout: lanes 0-15 hold K=0,1; lanes 16-31 K=2,3
    v2f a0 = *(const v2f*)(pa0 + ko);
    v2f a1 = *(const v2f*)(pa1 + ko);
    v2f b0 = *(const v2f*)(pb0 + ko);   // B(k,n)=Z[n][k]: identical per-lane pattern
    v2f b1 = *(const v2f*)(pb1 + ko);
    c00 = __builtin_amdgcn_wmma_f32_16x16x4_f32(false, a0, false, b0, (short)0, c00, false, false);
    c01 = __builtin_amdgcn_wmma_f32_16x16x4_f32(false, a0, false, b1, (short)0, c01, false, false);
    c10 = __builtin_amdgcn_wmma_f32_16x16x4_f32(false, a1, false, b0, (short)0, c10, false, false);
    c11 = __builtin_amdgcn_wmma_f32_16x16x4_f32(false, a1, false, b1, (short)0, c11, false, false);
  }
  float lmax = 0.0f;
  auto wr = [&](v8f cc, int rbase, int cbase) {
#pragma unroll
    for (int r = 0; r < 8; ++r) {
      int row = rbase + hi * 8 + r;       // C/D layout: VGPR r -> M = r + 8*hi, N = lane%16
      int col = cbase + m;
      float s = fmaxf(cc[r], 1e-7f);
      float cost = 1.0f - s;
      C[(size_t)row * NN + col] = cost;
      lmax = fmaxf(lmax, cost);
    }
  };
  wr(c00, rb, cb); wr(c01, rb, cb + 16); wr(c10, rb + 16, cb); wr(c11, rb + 16, cb + 16);
  __shared__ float sm[256];
  sm[threadIdx.x] = lmax; __syncthreads();
  for (int s = 128; s > 0; s >>= 1) {
    if (threadIdx.x < s) sm[threadIdx.x] = fmaxf(sm[threadIdx.x], sm[threadIdx.x + s]);
    __syncthreads();
  }
  if (threadIdx.x == 0) atomicMax(S + S_CMAX, __float_as_uint(sm[0]));
}

__global__ __launch_bounds__(256) void k_fixdiag(float* C, const unsigned* S) {
  int i = blockIdx.x * 256 + threadIdx.x;
  C[(size_t)i * NN + i] = __uint_as_float(S[S_CMAX]);
}

// ---------------- Sinkhorn iteration pieces ----------------
__global__ void k_iter_reset(unsigned* S) {
  if (threadIdx.x == 0) {
    S[S_DONE] = S[S_DONE] | S[S_BAD] | S[S_NONF];
    S[S_MAXU] = 0u; S[S_MAXV] = 0u; S[S_NONF] = 0u; S[S_BAD] = 0u;
  }
}

// y_i = sum_j exp((f_i+g_j-C_ij)/eps) v_j ; un_i = (a/(y+1e-16))^f1 * exp(-f_i/(eps+rel1))
__global__ __launch_bounds__(256) void k_matvec_rows(const float* __restrict__ C,
                                                     const float* __restrict__ g,
                                                     const float* __restrict__ v,
                                                     const float* __restrict__ f,
                                                     float* __restrict__ un,
                                                     unsigned* S) {
  int i = blockIdx.x, t = threadIdx.x;
  const float4* Crow = (const float4*)(C + (size_t)i * NN);
  const float4* g4 = (const float4*)g;
  const float4* v4 = (const float4*)v;
  float fi = f[i];
  float acc = 0.0f;
  for (int j4 = t; j4 < NN / 4; j4 += 256) {
    float4 c = Crow[j4]; float4 gg = g4[j4]; float4 vv = v4[j4];
    acc += __expf((fi + gg.x - c.x) * INV_EPS) * vv.x;
    acc += __expf((fi + gg.y - c.y) * INV_EPS) * vv.y;
    acc += __expf((fi + gg.z - c.z) * INV_EPS) * vv.z;
    acc += __expf((fi + gg.w - c.w) * INV_EPS) * vv.w;
  }
  __shared__ float sm[256];
  sm[t] = acc; __syncthreads();
  for (int s = 128; s > 0; s >>= 1) { if (t < s) sm[t] += sm[t + s]; __syncthreads(); }
  if (t == 0) {
    float y = sm[0];
    float uni = __powf(AW / (y + 1e-16f), F1C) * __expf(-fi / EPSREL);
    un[i] = uni;
    atomicMax(S + S_MAXU, __float_as_uint(fmaxf(uni, 0.0f)));
    if (!__builtin_isfinite(uni)) atomicOr(S + S_NONF, 1u);
  }
}

// partial column sums: spart[chunk][j] = sum_{i in chunk} exp((f_i+g_j-C_ij)/eps) un_i
__global__ __launch_bounds__(256) void k_matvec_cols_part(const float* __restrict__ C,
                                                          const float* __restrict__ f,
                                                          const float* __restrict__ g,
                                                          const float* __restrict__ un,
                                                          float* __restrict__ spart) {
  int j  = blockIdx.x * 256 + threadIdx.x;
  int i0 = blockIdx.y * (NN / 32);
  float gj = g[j];
  float acc = 0.0f;
  for (int i = i0; i < i0 + NN / 32; ++i) {
    float w = (f[i] + gj - C[(size_t)i * NN + j]) * INV_EPS;
    acc += __expf(w) * un[i];
  }
  spart[(size_t)blockIdx.y * NN + j] = acc;
}

// mode 0: vn_j from s_j (+max/finite flags). mode 1: zero-check on s_j -> bad.
__global__ __launch_bounds__(256) void k_cols_reduce(const float* __restrict__ spart,
                                                     const float* __restrict__ g,
                                                     float* __restrict__ vn,
                                                     unsigned* S, int mode) {
  int j = blockIdx.x * 256 + threadIdx.x;
  float s = 0.0f;
#pragma unroll
  for (int p = 0; p < 32; ++p) s += spart[(size_t)p * NN + j];
  if (mode == 0) {
    float vnj = __powf(AW / (s + 1e-16f), F1C) * __expf(-g[j] / EPSREL);
    vn[j] = vnj;
    atomicMax(S + S_MAXV, __float_as_uint(fmaxf(vnj, 0.0f)));
    if (!__builtin_isfinite(vnj)) atomicOr(S + S_NONF, 1u);
  } else {
    if (s == 0.0f) atomicOr(S + S_BAD, 1u);
  }
}

__global__ __launch_bounds__(256) void k_update_fg(float* f, float* g, float* vn,
                                                   const unsigned* S) {
  float maxu = __uint_as_float(S[S_MAXU]);
  float maxv = __uint_as_float(S[S_MAXV]);
  int done  = (int)S[S_DONE];
  int cond1 = (maxu > TAUC) || (maxv > TAUC);
  int j = blockIdx.x * 256 + threadIdx.x;
  if (cond1) {
    if (!done) {
      f[j] += EPSC * __logf(maxu);
      g[j] += EPSC * __logf(maxv);
    }
    vn[j] = 1.0f;
  }
}

__global__ __launch_bounds__(256) void k_commit(float* u, float* v,
                                                const float* un, const float* vn,
                                                const unsigned* S) {
  int keep = (int)(S[S_DONE] | S[S_BAD] | S[S_NONF]);
  int j = blockIdx.x * 256 + threadIdx.x;
  if (!keep) { u[j] = un[j]; v[j] = vn[j]; }
}

// ---------------- final loss ----------------
__global__ __launch_bounds__(256) void k_final_prep(const float* u, const float* v,
                                                    const float* f, const float* g,
                                                    float* logu, float* logv) {
  int j = blockIdx.x * 256 + threadIdx.x;
  logu[j] = f[j] * INV_EPS + __logf(u[j]);
  logv[j] = g[j] * INV_EPS + __logf(v[j]);
}

__global__ __launch_bounds__(256) void k_row_lse(const float* __restrict__ C,
                                                 const float* __restrict__ logu,
                                                 const float* __restrict__ logv,
                                                 const float* __restrict__ u,
                                                 float* __restrict__ kl_arr,
                                                 float* __restrict__ t_arr) {
  int i = blockIdx.x, t = threadIdx.x;
  const float* Crow = C + (size_t)i * NN;
  __shared__ float sm[256];
  float lm = -INFINITY;
  for (int j = t; j < NN; j += 256) lm = fmaxf(lm, logv[j] - Crow[j] * INV_EPS);
  sm[t] = lm; __syncthreads();
  for (int s = 128; s > 0; s >>= 1) { if (t < s) sm[t] = fmaxf(sm[t], sm[t + s]); __syncthreads(); }
  float m = sm[0];
  __syncthreads();
  float ls = 0.0f;
  for (int j = t; j < NN; j += 256) ls += __expf(logv[j] - Crow[j] * INV_EPS - m);
  sm[t] = ls; __syncthreads();
  for (int s = 128; s > 0; s >>= 1) { if (t < s) sm[t] += sm[t + s]; __syncthreads(); }
  if (t == 0) {
    int p = i ^ 1;
    float xp  = logv[p] - Crow[p] * INV_EPS;
    float lse = m + __logf(sm[0]);           // row-lse of (logv_j - C_ij/eps); logu_i cancels
    kl_arr[i] = -(xp - lse);
    float Psel = __expf(logu[i] + xp);       // P[i, p]
    float t1 = -__powf(Psel / u[i], QC) / QC;
    float t2 =  __powf(LAMC / u[p], QC) / QC;
    t_arr[i] = t1 + t2;
  }
}

__global__ __launch_bounds__(256) void k_final_reduce(const float* kl_arr,
                                                      const float* t_arr,
                                                      float* out) {
  int t = threadIdx.x;
  float sk = 0.0f, st = 0.0f;
  for (int i = t; i < NN; i += 256) { sk += kl_arr[i]; st += t_arr[i]; }
  __shared__ float a[256], b[256];
  a[t] = sk; b[t] = st; __syncthreads();
  for (int s = 128; s > 0; s >>= 1) {
    if (t < s) { a[t] += a[t + s]; b[t] += b[t + s]; }
    __syncthreads();
  }
  if (t == 0) out[0] = R1C * (a[0] / (float)NN) + R2C * (b[0] / (float)NN);
}

extern "C" void kernel_launch(void* const* d_in, const int* in_sizes, int n_in,
                              void* d_out, int out_size, void* d_ws, size_t ws_size,
                              hipStream_t stream) {
  (void)in_sizes; (void)n_in; (void)out_size; (void)ws_size;
  const float* z = (const float*)d_in[0];

  // workspace layout (floats): C[N*N], ZN[N*D], 10 vectors of N, SPART[32*N], flags
  float* W    = (float*)d_ws;
  float* C    = W;
  float* ZN   = C  + (size_t)NN * NN;
  float* U    = ZN + (size_t)NN * DD;
  float* V    = U  + NN;
  float* F    = V  + NN;
  float* G    = F  + NN;
  float* UNb  = G  + NN;
  float* VNb  = UNb + NN;
  float* LOGU = VNb + NN;
  float* LOGV = LOGU + NN;
  float* KL   = LOGV + NN;
  float* T    = KL + NN;
  float* SPART = T + NN;
  unsigned* S = (unsigned*)(SPART + (size_t)32 * NN);

  k_init<<<NN / 256, 256, 0, stream>>>(U, V, F, G, S);
  k_rownorm<<<NN, 256, 0, stream>>>(z, ZN);
  k_gemm_cost<<<dim3(NN / 128, NN / 64), 256, 0, stream>>>(ZN, C, S);
  k_fixdiag<<<NN / 256, 256, 0, stream>>>(C, S);

  for (int it = 0; it < 10; ++it) {
    k_iter_reset<<<1, 32, 0, stream>>>(S);
    k_matvec_rows<<<NN, 256, 0, stream>>>(C, G, V, F, UNb, S);
    k_matvec_cols_part<<<dim3(NN / 256, 32), 256, 0, stream>>>(C, F, G, UNb, SPART);
    k_cols_reduce<<<NN / 256, 256, 0, stream>>>(SPART, G, VNb, S, 0);
    k_update_fg<<<NN / 256, 256, 0, stream>>>(F, G, VNb, S);
    k_matvec_cols_part<<<dim3(NN / 256, 32), 256, 0, stream>>>(C, F, G, UNb, SPART);
    k_cols_reduce<<<NN / 256, 256, 0, stream>>>(SPART, G, VNb, S, 1);
    k_commit<<<NN / 256, 256, 0, stream>>>(U, V, UNb, VNb, S);
  }

  k_final_prep<<<NN / 256, 256, 0, stream>>>(U, V, F, G, LOGU, LOGV);
  k_row_lse<<<NN, 256, 0, stream>>>(C, LOGU, LOGV, U, KL, T);
  k_final_reduce<<<1, 256, 0, stream>>>(KL, T, (float*)d_out);
}